// QuantumQuanvolutionFilter_65481071402114
// MI455X (gfx1250) — compile-verified
//
#include <hip/hip_runtime.h>
#include <hip/hip_bf16.h>

// Quantum circuit expval collapses analytically:
//   RZ layers = diagonal unit phases (don't change |amp|^2)
//   CNOT chain = basis-state permutation that never touches bit 0
//   => expval Z0 = (cos^2(x0/2) - sin^2(x0/2)) * prod_w (cos^2+sin^2)
// Pure bandwidth-bound elementwise kernel: 16MB read + 4MB write ~ 0.9us @ 23.3TB/s.

typedef float v4f __attribute__((ext_vector_type(4)));

__global__ __launch_bounds__(256) void qz0_expval_kernel(
    const v4f* __restrict__ x, float* __restrict__ out, int n) {
  int i = blockIdx.x * blockDim.x + threadIdx.x;
  if (i >= n) return;

  // One coalesced 128-bit load per thread (whole row x[b,0..3]),
  // non-temporal: read-once data, no reuse, keep L2 clean.
  v4f v = __builtin_nontemporal_load(x + i);

  float s0, c0, s1, c1, s2, c2, s3, c3;
  __sincosf(0.5f * v.x, &s0, &c0);   // hardware v_sin_f32 / v_cos_f32 (trans ops)
  __sincosf(0.5f * v.y, &s1, &c1);
  __sincosf(0.5f * v.z, &s2, &c2);
  __sincosf(0.5f * v.w, &s3, &c3);

  // Track the reference's fp32 product structure: signed term on wire 0,
  // (cos^2 + sin^2) ~ 1 normalization factors for wires 1..3.
  float z  = c0 * c0 - s0 * s0;      // cos^2(x0/2) - sin^2(x0/2) = cos(x0)
  float n1 = c1 * c1 + s1 * s1;
  float n2 = c2 * c2 + s2 * s2;
  float n3 = c3 * c3 + s3 * s3;

  __builtin_nontemporal_store(z * n1 * n2 * n3, out + i);
}

extern "C" void kernel_launch(void* const* d_in, const int* in_sizes, int n_in,
                              void* d_out, int out_size, void* d_ws, size_t ws_size,
                              hipStream_t stream) {
  (void)in_sizes; (void)n_in; (void)d_ws; (void)ws_size;
  const v4f* x = (const v4f*)d_in[0];   // (BATCH, 4) float32, row = float4
  // d_in[1] (weight) provably does not affect the output; unused.
  float* out = (float*)d_out;           // (BATCH,) float32

  const int n = out_size;               // BATCH = 1048576
  const int block = 256;                // 8 wave32 waves per workgroup
  const int grid = (n + block - 1) / block;
  qz0_expval_kernel<<<grid, block, 0, stream>>>(x, out, n);
}